// UniRNN_40544491274322
// MI455X (gfx1250) — compile-verified
//
#include <hip/hip_runtime.h>

typedef float v2f __attribute__((ext_vector_type(2)));
typedef float v8f __attribute__((ext_vector_type(8)));
typedef unsigned int v4u __attribute__((ext_vector_type(4)));
typedef int v4i __attribute__((ext_vector_type(4)));
typedef int v8i __attribute__((ext_vector_type(8)));

#define B_  64
#define T_  512
#define E_  256
#define H_  512
#define G3_ 1536
#define NWG_ 32          // persistent workgroups in the scan (one per WGP)
#define CPW_ 16          // h-columns owned per workgroup
#define BSTRIDE 514      // padded K-stride for LDS rec_kernel tiles
#define HSTRIDE 516      // h row stride in LDS produced by TDM padding (512 + 2/256dw)
#define BS_FLOATS (3 * CPW_ * BSTRIDE)   // 24672 floats = 98688 B
#define HL_FLOATS (B_ * HSTRIDE)         // 33024 floats = 132096 B

__device__ __forceinline__ float sigmoidf_(float x) {
  return 1.0f / (1.0f + __expf(-x));
}

// ---------------------------------------------------------------------------
// Kernel 1: xp[b,t,:] = emb[tokens[b,t]] @ kernel + bias[0]
// One wave computes one 16x16 output tile with V_WMMA_F32_16X16X4_F32, K=256.
// ---------------------------------------------------------------------------
__global__ void __launch_bounds__(256) xp_gemm(const int* __restrict__ tokens,
                                               const float* __restrict__ emb,
                                               const float* __restrict__ Wk,
                                               const float* __restrict__ bias,
                                               float* __restrict__ xp)
{
  const int lane  = threadIdx.x & 31;
  const int wgl   = blockIdx.x * (blockDim.x >> 5) + (threadIdx.x >> 5);
  const int mTile = wgl % ((B_ * T_) / 16);   // 0..2047
  const int nTile = wgl / ((B_ * T_) / 16);   // 0..95 (waves in a block share B frag)
  const int row16 = lane & 15;
  const int kofs  = (lane >> 4) << 1;         // K pair: 0 or 2
  const int m0    = mTile * 16;
  const int col   = nTile * 16 + row16;

  const int tok = tokens[m0 + row16];
  const float* abase = emb + (long)tok * E_;
  const float b0 = bias[col];                 // bias[0] row

  v8f acc = {};
#pragma unroll 4
  for (int k = 0; k < E_; k += 4) {
    const int k0 = k + kofs;
    v2f a = *reinterpret_cast<const v2f*>(abase + k0);
    v2f b;
    b.x = Wk[k0 * G3_ + col];
    b.y = Wk[(k0 + 1) * G3_ + col];
    acc = __builtin_amdgcn_wmma_f32_16x16x4_f32(false, a, false, b,
                                                (short)0, acc, false, false);
  }
  const int rbase = m0 + ((lane >> 4) << 3);  // C/D: vgpr i -> row rbase+i
#pragma unroll
  for (int i = 0; i < 8; ++i)
    xp[(long)(rbase + i) * G3_ + col] = acc[i] + b0;
}

// ---------------------------------------------------------------------------
// Kernel 2: persistent GRU scan. 32 WGs x 128 threads (4 waves). WG wg owns h
// columns [wg*16, wg*16+16); wave w owns batch rows [16w, 16w+16). Per step:
//   - wave 0 TDM-loads the full h read-buffer (64x512 f32) into LDS with
//     padding (stride 516) -> s_wait_tensorcnt -> barrier
//   - each wave runs 3 WMMA f32 accumulator chains (z/r/h) from LDS
//   - gate math, masked update, store h_new tile to the global write buffer
//   - grid barrier via device-scope atomic counter
// ---------------------------------------------------------------------------
__global__ void __launch_bounds__(128) gru_scan(const int* __restrict__ tokens,
                                                const float* __restrict__ rk,
                                                const float* __restrict__ bias,
                                                const float* __restrict__ xp,
                                                float* __restrict__ hg,
                                                unsigned* __restrict__ counter)
{
  extern __shared__ float smem[];
  float* Bs   = smem;                   // [3*CPW_][BSTRIDE]
  float* hLds = smem + BS_FLOATS;       // [B_][HSTRIDE] (TDM-padded)

  const int wg   = blockIdx.x;          // 0..31
  const int tid  = threadIdx.x;
  const int lane = tid & 31;
  const int wave = tid >> 5;            // 0..3

  // Stage rec_kernel block transposed: Bs[g*16+ci][k] = rk[k][g*H + wg*16+ci]
  for (int idx = tid; idx < 3 * CPW_ * H_; idx += 128) {
    const int k  = idx & (H_ - 1);
    const int gc = idx >> 9;            // 0..47
    const int g  = gc >> 4;
    const int ci = gc & 15;
    Bs[gc * BSTRIDE + k] = rk[k * G3_ + g * H_ + wg * CPW_ + ci];
  }
  __syncthreads();

  const int m0    = wave * 16;          // batch-row tile
  const int row16 = lane & 15;
  const int kofs  = (lane >> 4) << 1;
  const int col   = wg * CPW_ + row16;  // absolute h column

  const float bz1 = bias[G3_ + col];    // bias[1] (recurrent), reset_after
  const float br1 = bias[G3_ + H_ + col];
  const float bh1 = bias[G3_ + 2 * H_ + col];

  const float* BsZ = &Bs[(0 * CPW_ + row16) * BSTRIDE];
  const float* BsR = &Bs[(1 * CPW_ + row16) * BSTRIDE];
  const float* BsH = &Bs[(2 * CPW_ + row16) * BSTRIDE];
  const float* hA  = &hLds[(m0 + row16) * HSTRIDE];   // A row; +2 after k=256
  const int rbase  = m0 + ((lane >> 4) << 3);

  const unsigned ldsOff =
      (unsigned)(unsigned long long)(unsigned long)(uintptr_t)hLds;

  v8f hreg = {};                        // this wave's h tile (carried)
  for (int t = 0; t < T_; ++t) {
    // ---- TDM: global h read-buffer -> LDS (wave 0 issues, all wait) ----
    if (wave == 0) {
      const unsigned long long ga =
          (unsigned long long)(uintptr_t)(hg + (size_t)(t & 1) * B_ * H_);
      v4u g0;
      g0[0] = 1u;                                  // count=1, user descriptor
      g0[1] = ldsOff;                              // lds_addr
      g0[2] = (unsigned)(ga & 0xFFFFFFFFull);      // global_addr[31:0]
      g0[3] = (unsigned)((ga >> 32) & 0x01FFFFFFull) | (2u << 30); // type=2
      v8i g1;
      g1[0] = (int)((2u << 16)          // data_size = 4B
                    | (1u << 20)        // pad_enable
                    | (7u << 22)        // pad_interval = 256 dwords
                    | (1u << 25));      // pad_amount  = 2 dwords
      g1[1] = (int)((32768u & 0xFFFFu) << 16);  // tensor_dim0 lo16 (=32768)
      g1[2] = (int)(1u << 16);                  // dim0 hi=0, tensor_dim1=1
      g1[3] = (int)(32768u << 16);              // dim1 hi=0, tile_dim0=32768
      g1[4] = 1;                                // tile_dim1=1, tile_dim2=0
      g1[5] = 32768;                            // tensor_dim0_stride lo
      g1[6] = 0;
      g1[7] = 0;
      v4i zz4 = {0, 0, 0, 0};
      v8i zz8 = {0, 0, 0, 0, 0, 0, 0, 0};
      __builtin_amdgcn_tensor_load_to_lds(g0, g1, zz4, zz4, zz8, 0);
      __builtin_amdgcn_s_wait_tensorcnt(0);
    }
    __syncthreads();

    // ---- 3 WMMA chains: rp_{z,r,h}[16x16] = h[16x512] @ B[512x16] ----
    v8f az = {}, ar = {}, ah = {};
#pragma unroll 4
    for (int k = 0; k < 256; k += 4) {          // first K half (no pad skew)
      const int k0 = k + kofs;
      v2f a  = *reinterpret_cast<const v2f*>(hA + k0);
      v2f bz = *reinterpret_cast<const v2f*>(BsZ + k0);
      v2f br = *reinterpret_cast<const v2f*>(BsR + k0);
      v2f bh = *reinterpret_cast<const v2f*>(BsH + k0);
      az = __builtin_amdgcn_wmma_f32_16x16x4_f32(false, a, false, bz, (short)0, az, false, false);
      ar = __builtin_amdgcn_wmma_f32_16x16x4_f32(false, a, false, br, (short)0, ar, false, false);
      ah = __builtin_amdgcn_wmma_f32_16x16x4_f32(false, a, false, bh, (short)0, ah, false, false);
    }
#pragma unroll 4
    for (int k = 256; k < 512; k += 4) {        // second K half: +2 dword pad
      const int k0 = k + kofs;
      v2f a  = *reinterpret_cast<const v2f*>(hA + 2 + k0);
      v2f bz = *reinterpret_cast<const v2f*>(BsZ + k0);
      v2f br = *reinterpret_cast<const v2f*>(BsR + k0);
      v2f bh = *reinterpret_cast<const v2f*>(BsH + k0);
      az = __builtin_amdgcn_wmma_f32_16x16x4_f32(false, a, false, bz, (short)0, az, false, false);
      ar = __builtin_amdgcn_wmma_f32_16x16x4_f32(false, a, false, br, (short)0, ar, false, false);
      ah = __builtin_amdgcn_wmma_f32_16x16x4_f32(false, a, false, bh, (short)0, ah, false, false);
    }

    // ---- gates + masked update; store tile to write buffer ----
    float* hw = hg + (size_t)((t + 1) & 1) * B_ * H_;
#pragma unroll
    for (int i = 0; i < 8; ++i) {
      const int  b  = rbase + i;
      const long xb = ((long)b * T_ + t) * G3_;
      const float xz = xp[xb + col];
      const float xr = xp[xb + H_ + col];
      const float xh = xp[xb + 2 * H_ + col];
      const float z  = sigmoidf_(xz + az[i] + bz1);
      const float r  = sigmoidf_(xr + ar[i] + br1);
      const float hh = tanhf(xh + r * (ah[i] + bh1));
      const float hn = z * hreg[i] + (1.0f - z) * hh;
      const bool  m  = tokens[b * T_ + t] != 0;
      hreg[i] = m ? hn : hreg[i];
      hw[b * H_ + col] = hreg[i];
    }

    // ---- grid-wide barrier: release stores, arrive, spin, acquire ----
    __threadfence();
    __syncthreads();
    if (tid == 0) {
      __hip_atomic_fetch_add(counter, 1u, __ATOMIC_RELEASE, __HIP_MEMORY_SCOPE_AGENT);
      const unsigned target = (unsigned)(NWG_ * (t + 1));
      while (__hip_atomic_load(counter, __ATOMIC_ACQUIRE, __HIP_MEMORY_SCOPE_AGENT) < target) {
        __builtin_amdgcn_s_sleep(1);
      }
    }
    __syncthreads();
    __threadfence();
  }
  // Final state resides in hg buffer 0 (T even).
}

// ---------------------------------------------------------------------------
// Kernel 3: logits = hT @ dense_w + dense_b   (64x512 @ 512x2 — trivial)
// ---------------------------------------------------------------------------
__global__ void dense_head(const float* __restrict__ hg,
                           const float* __restrict__ dw,
                           const float* __restrict__ db,
                           float* __restrict__ out)
{
  const int tid = threadIdx.x;
  if (tid < B_ * 2) {
    const int b = tid >> 1, c = tid & 1;
    float s = db[c];
    for (int k = 0; k < H_; ++k)
      s = fmaf(hg[b * H_ + k], dw[k * 2 + c], s);
    out[tid] = s;
  }
}

extern "C" void kernel_launch(void* const* d_in, const int* in_sizes, int n_in,
                              void* d_out, int out_size, void* d_ws, size_t ws_size,
                              hipStream_t stream)
{
  (void)in_sizes; (void)n_in; (void)out_size; (void)ws_size;
  const int*   tokens = (const int*)d_in[0];
  const float* emb    = (const float*)d_in[1];
  const float* Wk     = (const float*)d_in[2];
  const float* rk     = (const float*)d_in[3];
  const float* bias   = (const float*)d_in[4];
  const float* dw     = (const float*)d_in[5];
  const float* db     = (const float*)d_in[6];
  float* out = (float*)d_out;

  char* ws = (char*)d_ws;
  const size_t xpBytes = (size_t)B_ * T_ * G3_ * sizeof(float);   // 192 MB
  const size_t hBytes  = 2ull * B_ * H_ * sizeof(float);          // 256 KB
  float*    xp      = (float*)ws;
  float*    hg      = (float*)(ws + xpBytes);
  unsigned* counter = (unsigned*)(ws + xpBytes + hBytes);

  // zero h (both buffers) + barrier counter each launch (graph-capture safe)
  (void)hipMemsetAsync(hg, 0, hBytes + 256, stream);

  const int totalWaves = (B_ * T_ / 16) * (G3_ / 16);              // 196608
  xp_gemm<<<dim3(totalWaves / 8), dim3(256), 0, stream>>>(tokens, emb, Wk, bias, xp);

  const size_t ldsB = (size_t)(BS_FLOATS + HL_FLOATS) * sizeof(float); // ~225 KB
  (void)hipFuncSetAttribute((const void*)gru_scan,
                            hipFuncAttributeMaxDynamicSharedMemorySize, (int)ldsB);
  gru_scan<<<dim3(NWG_), dim3(128), ldsB, stream>>>(tokens, rk, bias, xp, hg, counter);

  dense_head<<<dim3(1), dim3(128), 0, stream>>>(hg, dw, db, out);
}